// AT_37821482009426
// MI455X (gfx1250) — compile-verified
//
#include <hip/hip_runtime.h>
#include <math.h>

// SNN leaky-integrate-fire over (batch=1024, h=224, T=224), roll(time, 57).
// One workgroup per batch element; 224 threads (7 full wave32s).
//
// LDS column chunks (source-column index s):
//   A: s in [164,224), stride 60  -> scan t=0..56   (s=167..223) + t=221..223 (s=164..166, via regs)
//   B: s in [0,112),   stride 116 -> scan t=57..168
//   C: s in [112,164), stride 52  -> scan t=169..220 (reuses A's buffer)

#define BETA 0.95f

#define S_A 60
#define S_B 116
#define S_C 52
#define OFF_B (224 * S_A)            // 13440 floats
#define OFF_T (OFF_B + 224 * S_B)    // 39424
#define OFF_W (OFF_T + 224)          // 39648
#define LDS_FLOATS (OFF_W + 224)     // 39872 floats = 159,488 bytes

typedef float v2f __attribute__((ext_vector_type(2)));
typedef float v8f __attribute__((ext_vector_type(8)));
typedef int i4 __attribute__((vector_size(16)));
typedef __attribute__((address_space(1))) i4 gi4;   // global int4
typedef __attribute__((address_space(3))) i4 li4;   // LDS int4

#if __has_builtin(__builtin_amdgcn_global_load_async_to_lds_b128)
#define CP16(g, l) __builtin_amdgcn_global_load_async_to_lds_b128((gi4*)(g), (li4*)(l), 0, 0)
#if __has_builtin(__builtin_amdgcn_s_wait_asynccnt)
#define CPWAIT() __builtin_amdgcn_s_wait_asynccnt(0)
#else
#define CPWAIT() asm volatile("s_wait_asynccnt 0" ::: "memory")
#endif
#else
// Fallback: synchronous global->VGPR->LDS copy (b128 both sides).
#define CP16(g, l) (*(float4*)(l) = *(const float4*)(g))
#define CPWAIT() ((void)0)
#endif

// Copy a 224-row x (4*W4)-column chunk (global row stride 224 floats) into LDS
// with row stride S floats. 224*W4 16-byte packets, exactly W4 per thread.
template <int W4, int S>
__device__ __forceinline__ void stage_chunk(const float* __restrict__ gbase,
                                            float* lbase, int tid) {
  for (int r = 0; r < W4; ++r) {
    const int idx = tid + r * 224;
    const int hh = idx / W4;
    const int q = idx - hh * W4;
    CP16(gbase + hh * 224 + q * 4, lbase + hh * S + q * 4);
  }
}

// Thresholds for 16 source columns starting at tc0 of a chunk:
// T[sbase+tc0+n] = clip(3 + 2*tanh(sum_h w[h]*x[h][tc0+n]), 1, 5)
// Whole wave participates (EXEC all ones for WMMA); lanes < nvalid write.
template <int S>
__device__ __forceinline__ void dot_tile(const float* buf, const float* wl, float* Tl,
                                         int tc0, int nvalid, int sbase, int lane) {
#if __has_builtin(__builtin_amdgcn_wmma_f32_16x16x4_f32)
  v8f acc = {0.f, 0.f, 0.f, 0.f, 0.f, 0.f, 0.f, 0.f};
  const int kof = (lane >> 4) << 1;      // K sub-offset per half-wave (0 or 2)
  const int col = tc0 + (lane & 15);     // tile column for B fragment
  #pragma unroll 4
  for (int k0 = 0; k0 < 224; k0 += 4) {
    const int kr = k0 + kof;
    v2f af;                               // A = w replicated across all 16 rows
    af.x = wl[kr];
    af.y = wl[kr + 1];
    v2f bf;                               // B = x[kr..kr+1][col]
    bf.x = buf[kr * S + col];
    bf.y = buf[(kr + 1) * S + col];
    acc = __builtin_amdgcn_wmma_f32_16x16x4_f32(false, af, false, bf,
                                                (short)0, acc, false, false);
  }
  if (lane < nvalid) {                    // every D row equals the dot; lane n has d[tc0+n]
    const float thr = 3.0f + 2.0f * tanhf(acc[0]);
    Tl[sbase + tc0 + lane] = fminf(5.0f, fmaxf(1.0f, thr));
  }
#else
  if (lane < 16) {
    const int c = tc0 + lane;
    float d = 0.0f;
    for (int k = 0; k < 224; ++k) d = fmaf(wl[k], buf[k * S + c], d);
    if (lane < nvalid) {
      const float thr = 3.0f + 2.0f * tanhf(d);
      Tl[sbase + tc0 + lane] = fminf(5.0f, fmaxf(1.0f, thr));
    }
  }
#endif
}

// Leaky scan over n consecutive source columns j0..j0+n-1 of a chunk,
// emitting spikes at output times t..t+n-1. Returns updated t.
template <int S>
__device__ __forceinline__ int scan_seg(const float* buf, const float* Tl,
                                        float* __restrict__ orow,
                                        int j0, int n, int sbase, int h, int t,
                                        float& mem) {
  #pragma unroll 4
  for (int i = 0; i < n; ++i) {
    const float x = buf[h * S + j0 + i];
    const float Tt = Tl[sbase + j0 + i];
    mem = BETA * mem + x - ((mem > Tt) ? Tt : 0.0f);   // reset uses OLD mem
    orow[t + i] = (mem > Tt) ? 1.0f : 0.0f;
  }
  return t + n;
}

__global__ __launch_bounds__(224) void snn_leaky_kernel(const float* __restrict__ inp,
                                                        const float* __restrict__ w,
                                                        float* __restrict__ out) {
  __shared__ __align__(16) float lds[LDS_FLOATS];
  float* bufAC = lds;              // chunk A, later chunk C
  float* bufB = lds + OFF_B;       // chunk B
  float* Tl = lds + OFF_T;         // 224 thresholds, indexed by source column s
  float* wl = lds + OFF_W;         // 224 weights

  const int b = blockIdx.x;
  const int tid = threadIdx.x;     // 0..223
  const int wave = tid >> 5;
  const int lane = tid & 31;
  const float* gin = inp + (size_t)b * (224 * 224);

  // Async-stage w, chunk A (cols 164..223) and chunk B (cols 0..111).
  if (tid < 56) CP16(w + tid * 4, wl + tid * 4);
  stage_chunk<15, S_A>(gin + 164, bufAC, tid);
  stage_chunk<28, S_B>(gin + 0, bufB, tid);
  CPWAIT();
  __syncthreads();

  // Thresholds for chunks A and B: 11 WMMA tiles over 7 waves.
  for (int ti = wave; ti < 11; ti += 7) {
    if (ti < 4)
      dot_tile<S_A>(bufAC, wl, Tl, 16 * ti, (ti == 3) ? 12 : 16, 164, lane);
    else
      dot_tile<S_B>(bufB, wl, Tl, 16 * (ti - 4), 16, 0, lane);
  }
  __syncthreads();

  const int h = tid;
  float* orow = out + (size_t)b * (224 * 224) + h * 224;
  float mem = 0.0f;

  // Save source cols 164..166 (needed at t=221..223) before C overwrites bufAC.
  const float ta0 = bufAC[h * S_A + 0];
  const float ta1 = bufAC[h * S_A + 1];
  const float ta2 = bufAC[h * S_A + 2];

  // Phase 1: t = 0..56  <- source cols 167..223 (chunk A, j=3..59).
  int t = scan_seg<S_A>(bufAC, Tl, orow, 3, 57, 164, h, 0, mem);
  __syncthreads();  // everyone done reading bufAC

  // Prefetch chunk C (cols 112..163) into bufAC; hide behind the B scan.
  stage_chunk<13, S_C>(gin + 112, bufAC, tid);

  // Phase 2: t = 57..168 <- source cols 0..111 (chunk B).
  t = scan_seg<S_B>(bufB, Tl, orow, 0, 112, 0, h, t, mem);

  CPWAIT();
  __syncthreads();

  // Thresholds for chunk C (source cols 112..163): 4 tiles on waves 0..3.
  for (int ti = wave; ti < 4; ti += 7)
    dot_tile<S_C>(bufAC, wl, Tl, 16 * ti, (ti == 3) ? 4 : 16, 112, lane);
  __syncthreads();

  // Phase 3: t = 169..220 <- source cols 112..163 (chunk C).
  t = scan_seg<S_C>(bufAC, Tl, orow, 0, 52, 112, h, t, mem);

  // Phase 4: t = 221..223 <- source cols 164..166 (saved in registers).
  {
    const float xs0[3] = {ta0, ta1, ta2};
    #pragma unroll
    for (int i = 0; i < 3; ++i) {
      const float Tt = Tl[164 + i];
      mem = BETA * mem + xs0[i] - ((mem > Tt) ? Tt : 0.0f);
      orow[221 + i] = (mem > Tt) ? 1.0f : 0.0f;
    }
  }
}

extern "C" void kernel_launch(void* const* d_in, const int* in_sizes, int n_in,
                              void* d_out, int out_size, void* d_ws, size_t ws_size,
                              hipStream_t stream) {
  const float* inp = (const float*)d_in[0];
  const float* w = (const float*)d_in[1];
  float* out = (float*)d_out;
  const int batch = in_sizes[0] / (224 * 224);
  snn_leaky_kernel<<<dim3(batch), dim3(224), 0, stream>>>(inp, w, out);
}